// Gene2SubDifferCrossAttn_12154757448324
// MI455X (gfx1250) — compile-verified
//
#include <hip/hip_runtime.h>
#include <stdint.h>

typedef __attribute__((ext_vector_type(2))) float v2f;
typedef __attribute__((ext_vector_type(8))) float v8f;

#define B_   8
#define L_   8192     // P*G
#define S_   256
#define E_   32
#define H_   16
#define SUBE_ 768
#define ROWS_ 32
#define LAMBDA_INIT_ 0.2f    // 0.8 - 0.6*exp(0)
#define EPS_SM_ 1e-8f
#define EPS_LN_ 1e-5f

#if defined(__HIP_DEVICE_COMPILE__)
#define USE_ASYNC_LDS 1
#else
#define USE_ASYNC_LDS 0
#endif

__device__ __forceinline__ v8f wmma_f32(v2f a, v2f b, v8f c) {
  // D = A(16x4) * B(4x16) + C(16x16), fp32, wave32
  return __builtin_amdgcn_wmma_f32_16x16x4_f32(false, a, false, b, (short)0, c,
                                               false, false);
}

// ---------------------------------------------------------------------------
// Kernel 1: KV = key @ Wkv.T   -> split into K (first 32 cols) and V (last 32)
// One wave per 16x16 output tile; 192 f32 WMMA k-steps (K = 768).
// ---------------------------------------------------------------------------
__global__ void kv_kernel(const float* __restrict__ key,
                          const float* __restrict__ Wkv,
                          float* __restrict__ Kws,
                          float* __restrict__ Vws) {
  int idx  = blockIdx.x;
  int b    = idx >> 6;          // 64 tiles per batch (16 s-tiles * 4 n-tiles)
  int rem  = idx & 63;
  int st   = rem >> 2;
  int nt   = rem & 3;
  int lane = threadIdx.x;
  int l15  = lane & 15;
  int koff = (lane >> 4) << 1;  // lanes 16-31 hold K+2/K+3
  int mrow = st * 16 + l15;
  int ncol = nt * 16 + l15;

  const float* ap = key + ((size_t)b * S_ + mrow) * SUBE_ + koff;
  const float* bp = Wkv + (size_t)ncol * SUBE_ + koff;

  v8f c = {};
#pragma unroll 8
  for (int k0 = 0; k0 < SUBE_; k0 += 4) {
    v2f a;  a.x  = ap[k0]; a.y  = ap[k0 + 1];
    v2f bb; bb.x = bp[k0]; bb.y = bp[k0 + 1];
    c = wmma_f32(a, bb, c);
  }

  int radd = (lane >= 16) ? 8 : 0;
  // nt<2 -> K half, nt>=2 -> V half (block-uniform: no per-lane branch)
  float* dst = (nt < 2) ? Kws : Vws;
  int col = ((nt & 1) << 4) + l15;
  size_t base = ((size_t)b * S_ + st * 16 + radd) * E_ + col;
#pragma unroll
  for (int r = 0; r < 8; ++r)
    dst[base + (size_t)r * E_] = c[r];
}

// ---------------------------------------------------------------------------
// Kernel 2: fused differential cross-attention for a 32-row slab of one batch.
// 64 threads = 2 wave32s; each wave owns 16 rows for the WMMA GEMMs.
// LDS layout (byte offsets):
//   Ks 0 | Vs 32768 | Qs 65536 | Wqs 69632 | Wos 73728 |
//   sc0 77824 | sc1 110592 | lnws 143360 | kms 143488 | inv 144512
// ---------------------------------------------------------------------------
#define KS_OFF_   0u
#define VS_OFF_   ((unsigned)(S_ * E_ * 4))
#define SC0_OFF_  ((unsigned)((2 * S_ * E_ + ROWS_ * E_ + 2 * E_ * E_) * 4))

__global__ void attn_kernel(const float* __restrict__ query,
                            const unsigned char* __restrict__ qmask,
                            const unsigned char* __restrict__ kmask,
                            const float* __restrict__ Wq,
                            const float* __restrict__ Wout,
                            const float* __restrict__ lq1,
                            const float* __restrict__ lk1,
                            const float* __restrict__ lq2,
                            const float* __restrict__ lk2,
                            const float* __restrict__ lnw,
                            const float* __restrict__ Kws,
                            const float* __restrict__ Vws,
                            float* __restrict__ out,
                            float* __restrict__ diff_out) {
  extern __shared__ float smem[];
  float* Ks   = smem;                 // [S_][E_]
  float* Vs   = Ks  + S_ * E_;        // [S_][E_]
  float* Qs   = Vs  + S_ * E_;        // [ROWS_][E_] (later reused as attn_out)
  float* Wqs  = Qs  + ROWS_ * E_;     // [E_][E_]
  float* Wos  = Wqs + E_ * E_;        // [E_][E_]
  float* sc0  = Wos + E_ * E_;        // [ROWS_][S_] (gene -> scores h0 -> diff)
  float* sc1  = sc0 + ROWS_ * S_;     // [ROWS_][S_] (scores h1 -> exp h1)
  float* lnws = sc1 + ROWS_ * S_;     // [E_]
  float* kms  = lnws + E_;            // [S_]
  float* inv0 = kms + S_;             // [ROWS_]
  float* inv1 = inv0 + ROWS_;         // [ROWS_]

  int b    = blockIdx.x / (L_ / ROWS_);
  int tile = blockIdx.x % (L_ / ROWS_);
  int row0 = tile * ROWS_;
  int tid  = threadIdx.x;             // 0..63
  int lane = tid & 31;
  int wave = tid >> 5;
  int l15  = lane & 15;
  int koff = (lane >> 4) << 1;
  int radd = (lane >= 16) ? 8 : 0;
  int wr0  = wave * 16;

  // ---- stage K/V via CDNA5 async LDS loads (ASYNCcnt path) ----
  const float4* Kg = (const float4*)(Kws + (size_t)b * S_ * E_);
  const float4* Vg = (const float4*)(Vws + (size_t)b * S_ * E_);
#if USE_ASYNC_LDS
  for (int i = tid; i < S_ * E_ / 4; i += 64) {
    asm volatile("global_load_async_to_lds_b128 %0, %1, off"
                 :: "v"(KS_OFF_ + (unsigned)i * 16u), "v"(Kg + i) : "memory");
    asm volatile("global_load_async_to_lds_b128 %0, %1, off"
                 :: "v"(VS_OFF_ + (unsigned)i * 16u), "v"(Vg + i) : "memory");
  }
#else
  for (int i = tid; i < S_ * E_ / 4; i += 64) {
    ((float4*)Ks)[i] = Kg[i];
    ((float4*)Vs)[i] = Vg[i];
  }
#endif

  // ---- weights / gene / masks via regular loads (small) ----
  const float4* Wqg = (const float4*)Wq;
  const float4* Wog = (const float4*)Wout;
  for (int i = tid; i < E_ * E_ / 4; i += 64) {
    ((float4*)Wqs)[i] = Wqg[i];
    ((float4*)Wos)[i] = Wog[i];
  }
  const float4* Gg = (const float4*)(query + ((size_t)b * L_ + row0) * E_);
  for (int i = tid; i < ROWS_ * E_ / 4; i += 64)
    ((float4*)sc0)[i] = Gg[i];        // gene tile parked in sc0
  if (tid < E_) lnws[tid] = lnw[tid];
  for (int s = tid; s < S_; s += 64)
    kms[s] = kmask[(size_t)b * S_ + s] ? 1.0f : 0.0f;

#if USE_ASYNC_LDS
  asm volatile("s_wait_asynccnt 0" ::: "memory");
#endif
  __syncthreads();

  // ---- Q projection (pre-scaled by HEAD_D^-0.5 = 0.25) ----
  for (int o = tid; o < ROWS_ * E_; o += 64) {
    int r = o >> 5, e = o & 31;
    float acc = 0.f;
#pragma unroll
    for (int k = 0; k < E_; ++k) acc += sc0[r * E_ + k] * Wqs[e * E_ + k];
    Qs[o] = acc * 0.25f;
  }
  __syncthreads();

  // ---- scores for both heads via f32 WMMA; mask -> -1e20; park in LDS ----
  int arow = wr0 + l15;               // tile-local M row for this lane
  for (int h = 0; h < 2; ++h) {
    float* sc = h ? sc1 : sc0;
    for (int st = 0; st < 16; ++st) {
      v8f c = {};
#pragma unroll
      for (int ks = 0; ks < 4; ++ks) {
        int k0 = h * 16 + ks * 4 + koff;
        v2f a;  a.x  = Qs[arow * E_ + k0];
                a.y  = Qs[arow * E_ + k0 + 1];
        v2f bb; bb.x = Ks[(st * 16 + l15) * E_ + k0];
                bb.y = Ks[(st * 16 + l15) * E_ + k0 + 1];
        c = wmma_f32(a, bb, c);
      }
      int n = st * 16 + l15;
      float km = kms[n];
#pragma unroll
      for (int r = 0; r < 8; ++r) {
        int lrow = wr0 + r + radd;
        float qm = qmask[(size_t)b * L_ + row0 + lrow] ? 1.0f : 0.0f;
        float v = c[r];
        if (km * qm == 0.0f) v = -1e20f;
        sc[lrow * S_ + n] = v;
      }
    }
  }
  __syncthreads();

  // ---- softmax: wave0 handles head0 rows, wave1 handles head1 rows ----
  {
    int r = tid & 31;                 // one row per lane
    float* sc = wave ? sc1 : sc0;
    float* iv = wave ? inv1 : inv0;
    float m = -3.4e38f;
    for (int j = 0; j < S_; ++j) {
      int s = (r * 8 + j) & (S_ - 1); // bank-staggered row walk
      m = fmaxf(m, sc[r * S_ + s]);
    }
    float sum = 0.f;
    for (int j = 0; j < S_; ++j) {
      int s = (r * 8 + j) & (S_ - 1);
      float e = __expf(sc[r * S_ + s] - m);
      sc[r * S_ + s] = e;
      sum += e;
    }
    iv[r] = 1.f / (sum + EPS_SM_);
  }
  __syncthreads();

  // ---- lambda-difference, min-subtract, mask (rows on threads 0..31) ----
  if (tid < ROWS_) {
    float d1 = 0.f, d2 = 0.f;
#pragma unroll
    for (int i = 0; i < H_; ++i) { d1 += lq1[i] * lk1[i]; d2 += lq2[i] * lk2[i]; }
    float lam = __expf(d1) - __expf(d2) + LAMBDA_INIT_;

    int r = tid;
    int grow = row0 + r;
    float qm = qmask[(size_t)b * L_ + grow] ? 1.0f : 0.0f;
    float i0 = inv0[r], i1 = inv1[r];

    float dmin = 3.4e38f;
    for (int j = 0; j < S_; ++j) {
      int s = (r * 8 + j) & (S_ - 1);
      float d = sc0[r * S_ + s] * i0 - lam * (sc1[r * S_ + s] * i1);
      sc0[r * S_ + s] = d;
      dmin = fminf(dmin, d);
    }
    for (int j = 0; j < S_; ++j) {
      int s = (r * 8 + j) & (S_ - 1);
      float d = sc0[r * S_ + s] - dmin + 1e-20f;
      if (qm * kms[s] == 0.0f) d = 0.0f;
      sc0[r * S_ + s] = d;
    }
  }
  __syncthreads();

  // ---- stream diff tile (contiguous 32x256 block) straight out of LDS ----
  // Async b128 stores overlap with the diff @ V WMMA GEMM below; S_ENDPGM's
  // implicit wait-idle drains ASYNCcnt, and sc0 is never written again.
  {
    float4* Dg = (float4*)(diff_out + ((size_t)b * L_ + row0) * S_);
#if USE_ASYNC_LDS
    for (int i = tid; i < ROWS_ * S_ / 4; i += 64) {
      asm volatile("global_store_async_from_lds_b128 %0, %1, off"
                   :: "v"(Dg + i), "v"(SC0_OFF_ + (unsigned)i * 16u) : "memory");
    }
#else
    for (int i = tid; i < ROWS_ * S_ / 4; i += 64)
      Dg[i] = ((float4*)sc0)[i];
#endif
  }

  // ---- attn_out = diff(32x256) @ V(256x32) via f32 WMMA (K = 256) ----
  for (int nt = 0; nt < 2; ++nt) {
    v8f c = {};
    int ncol = nt * 16 + l15;
#pragma unroll 4
    for (int k0 = 0; k0 < S_; k0 += 4) {
      v2f a;  a.x  = sc0[arow * S_ + k0 + koff];
              a.y  = sc0[arow * S_ + k0 + koff + 1];
      v2f bb; bb.x = Vs[(k0 + koff) * E_ + ncol];
              bb.y = Vs[(k0 + koff + 1) * E_ + ncol];
      c = wmma_f32(a, bb, c);
    }
#pragma unroll
    for (int r = 0; r < 8; ++r) {
      int lrow = wr0 + r + radd;
      Qs[lrow * E_ + ncol] = c[r];    // reuse Qs as attn_out staging
    }
  }
  __syncthreads();

  // ---- fused RMSNorm * (1-lambda_init) then @ Wout.T ----
  if (tid < ROWS_) {
    int r = tid;
    int grow = row0 + r;
    float var = 0.f;
    float xr[E_];
#pragma unroll
    for (int e = 0; e < E_; ++e) { float x = Qs[r * E_ + e]; xr[e] = x; var += x * x; }
    var *= (1.0f / E_);
    float rs = rsqrtf(var + EPS_LN_) * (1.0f - LAMBDA_INIT_);
#pragma unroll
    for (int e = 0; e < E_; ++e) xr[e] *= rs * lnws[e];
    for (int eo = 0; eo < E_; ++eo) {
      float acc = 0.f;
#pragma unroll
      for (int e = 0; e < E_; ++e) acc += xr[e] * Wos[eo * E_ + e];
      out[((size_t)b * L_ + grow) * E_ + eo] = acc;
    }
  }
}

// ---------------------------------------------------------------------------
extern "C" void kernel_launch(void* const* d_in, const int* in_sizes, int n_in,
                              void* d_out, int out_size, void* d_ws, size_t ws_size,
                              hipStream_t stream) {
  (void)in_sizes; (void)n_in; (void)out_size; (void)ws_size;

  const float*         query = (const float*)d_in[0];
  const float*         key   = (const float*)d_in[1];
  const unsigned char* qmask = (const unsigned char*)d_in[2];
  const unsigned char* kmask = (const unsigned char*)d_in[3];
  const float*         Wq    = (const float*)d_in[4];
  const float*         Wkv   = (const float*)d_in[5];
  const float*         Wout  = (const float*)d_in[6];
  const float*         lq1   = (const float*)d_in[7];
  const float*         lk1   = (const float*)d_in[8];
  const float*         lq2   = (const float*)d_in[9];
  const float*         lk2   = (const float*)d_in[10];
  const float*         lnw   = (const float*)d_in[11];

  float* Kws = (float*)d_ws;                       // [B][S][E]
  float* Vws = Kws + (size_t)B_ * S_ * E_;         // [B][S][E]
  float* outp = (float*)d_out;                     // [B][L][E]
  float* diff = outp + (size_t)B_ * L_ * E_;       // [B][L][S]

  kv_kernel<<<dim3(B_ * 16 * 4), dim3(32), 0, stream>>>(key, Wkv, Kws, Vws);

  size_t smem_bytes = (size_t)(2 * S_ * E_ +        // Ks, Vs
                               ROWS_ * E_ +         // Qs
                               2 * E_ * E_ +        // Wqs, Wos
                               2 * ROWS_ * S_ +     // sc0, sc1
                               E_ + S_ +            // lnws, kms
                               2 * ROWS_) *         // inv0, inv1
                      sizeof(float);                // ~144.8 KB (<320 KB WGP LDS)

  attn_kernel<<<dim3(B_ * (L_ / ROWS_)), dim3(64), smem_bytes, stream>>>(
      query, qmask, kmask, Wq, Wout, lq1, lk1, lq2, lk2, lnw, Kws, Vws, outp, diff);
}